// GCN_47373489274965
// MI455X (gfx1250) — compile-verified
//
#include <hip/hip_runtime.h>
#include <hip/hip_bf16.h>

#define D 128
#define EPS 1e-5f

typedef __attribute__((ext_vector_type(2))) float v2f;
typedef __attribute__((ext_vector_type(8))) float v8f;

// ---------------------------------------------------------------------------
// utility: zero-fill
// ---------------------------------------------------------------------------
__global__ void gcn_zero_kernel(float* __restrict__ p, long long n) {
    long long i = (long long)blockIdx.x * blockDim.x + threadIdx.x;
    long long stride = (long long)gridDim.x * blockDim.x;
    for (; i < n; i += stride) p[i] = 0.0f;
}

// ---------------------------------------------------------------------------
// degree: deg[dst[e]] += 1 over real edges (self-loop +1 folded in later)
// ---------------------------------------------------------------------------
__global__ void gcn_deg_kernel(const int* __restrict__ dst, float* __restrict__ deg, int E) {
    int i = blockIdx.x * blockDim.x + threadIdx.x;
    if (i < E) atomicAdd(&deg[dst[i]], 1.0f);
}

__global__ void gcn_dinv_kernel(const float* __restrict__ deg, float* __restrict__ dinv, int N) {
    int i = blockIdx.x * blockDim.x + threadIdx.x;
    if (i < N) dinv[i] = rsqrtf(deg[i] + 1.0f);   // +1 = self loop; always > 0
}

// ---------------------------------------------------------------------------
// WMMA fp32 GEMM: out[M x 128] = A[M x 128] @ W^T  (+ bias), W row-major
// [out_feat, in_feat].  Block = 256 thr = 8 waves; block owns 32 rows
// (two 16-row tiles), wave w owns columns [16w, 16w+16).  Each wave keeps
// TWO accumulators so every B fragment (W^T) feeds 2 WMMAs -> halves the
// L2 re-read of W and doubles WMMA density per load.
// A staged in LDS: 32x128 f32 = 16 KB.
// ---------------------------------------------------------------------------
__global__ void gcn_gemm_wmma_kernel(const float* __restrict__ A,
                                     const float* __restrict__ W,
                                     const float* __restrict__ bias,
                                     float* __restrict__ out, int M) {
    __shared__ float ldsA[32 * D];
    const int row0 = blockIdx.x * 32;
    const int tid  = threadIdx.x;

    // stage 32x128 A tile (1024 float4's, 256 threads -> 4 each)
    for (int i4 = tid; i4 < (32 * D) / 4; i4 += 256) {
        int r = row0 + (i4 >> 5);                 // 32 float4 per row
        float4 v = make_float4(0.f, 0.f, 0.f, 0.f);
        if (r < M) v = ((const float4*)(A))[(long long)r * (D / 4) + (i4 & 31)];
        ((float4*)ldsA)[i4] = v;
    }
    __syncthreads();

    const int wave = tid >> 5;          // 0..7 -> column tile
    const int lane = tid & 31;
    const int half = lane >> 4;         // lane group 0 / 1
    const int mn   = lane & 15;         // row (A) or col (B/D) index
    const int col0 = wave * 16;

    v8f c0 = {};                        // rows row0      .. row0+15
    v8f c1 = {};                        // rows row0 + 16 .. row0+31
    #pragma unroll
    for (int k0 = 0; k0 < D; k0 += 4) {
        const int k = k0 + 2 * half;    // lanes 16-31 hold K+2 / K+3
        v2f b  = *(const v2f*)(W + (long long)(col0 + mn) * D + k); // B = W^T 4x16
        v2f a0 = *(const v2f*)(ldsA + mn * D + k);                  // tile 0 A frag
        v2f a1 = *(const v2f*)(ldsA + (mn + 16) * D + k);           // tile 1 A frag
        c0 = __builtin_amdgcn_wmma_f32_16x16x4_f32(false, a0, false, b,
                                                   (short)0, c0, false, false);
        c1 = __builtin_amdgcn_wmma_f32_16x16x4_f32(false, a1, false, b,
                                                   (short)0, c1, false, false);
    }

    const float bv = bias ? bias[col0 + mn] : 0.0f;
    #pragma unroll
    for (int j = 0; j < 8; ++j) {
        int m = row0 + j + 8 * half;    // D layout: VGPR j -> M=j (lanes 0-15) / j+8
        if (m < M)      out[(long long)m * D + col0 + mn]        = c0[j] + bv;
        if (m + 16 < M) out[(long long)(m + 16) * D + col0 + mn] = c1[j] + bv;
    }
}

// ---------------------------------------------------------------------------
// edge scatter: agg[dst] += t[src] * dinv[src]*dinv[dst]
// one wave32 per segment (E real edges then N self loops); lane -> float4.
// h/t/agg are L2-resident (25.6 MB each vs 192 MB L2) so this is an
// L2-atomic-throughput problem, not an HBM problem.
// ---------------------------------------------------------------------------
__global__ void gcn_scatter_kernel(const float* __restrict__ t,
                                   const int* __restrict__ src,
                                   const int* __restrict__ dst,
                                   const float* __restrict__ dinv,
                                   float* __restrict__ agg, int E, int N) {
    long long gw = ((long long)blockIdx.x * blockDim.x + threadIdx.x) >> 5;
    int lane = threadIdx.x & 31;
    long long total = (long long)E + N;
    if (gw >= total) return;
    int s, d;
    if (gw < E) { s = src[gw]; d = dst[gw]; }
    else        { s = d = (int)(gw - E); }
    float nrm = dinv[s] * dinv[d];
    const float4 v = *(const float4*)(t + (long long)s * D + lane * 4);
    float* p = agg + (long long)d * D + lane * 4;
    atomicAdd(p + 0, v.x * nrm);
    atomicAdd(p + 1, v.y * nrm);
    atomicAdd(p + 2, v.z * nrm);
    atomicAdd(p + 3, v.w * nrm);
}

// ---------------------------------------------------------------------------
// batchnorm column stats: per-column sum / sumsq over M rows
// (bias bc cancels exactly in (x - mean), so stats on raw agg are exact)
// ---------------------------------------------------------------------------
__global__ void gcn_bn_stats_kernel(const float* __restrict__ agg,
                                    float* __restrict__ sums,
                                    float* __restrict__ sqs, int M) {
    int col = threadIdx.x;                          // 128 threads
    int rows_per_block = (M + gridDim.x - 1) / gridDim.x;
    int r0 = blockIdx.x * rows_per_block;
    int r1 = r0 + rows_per_block; if (r1 > M) r1 = M;
    float s = 0.f, q = 0.f;
    for (int r = r0; r < r1; ++r) {
        float v = agg[(long long)r * D + col];
        s += v; q += v * v;
    }
    atomicAdd(&sums[col], s);
    atomicAdd(&sqs[col], q);
}

__global__ void gcn_bn_finalize_kernel(const float* __restrict__ sums,
                                       const float* __restrict__ sqs,
                                       const float* __restrict__ gamma,
                                       const float* __restrict__ beta,
                                       float* __restrict__ scale,
                                       float* __restrict__ shift, int M) {
    int col = threadIdx.x;
    if (col >= D) return;
    float inv_m = 1.0f / (float)M;
    float mean = sums[col] * inv_m;
    float var  = sqs[col] * inv_m - mean * mean;
    float sc   = gamma[col] * rsqrtf(var + EPS);
    scale[col] = sc;
    shift[col] = beta[col] - mean * sc;
}

// h = relu(agg*scale + shift) + h   (fused BN + ReLU + residual)
__global__ void gcn_bn_apply_kernel(const float* __restrict__ agg,
                                    const float* __restrict__ scale,
                                    const float* __restrict__ shift,
                                    float* __restrict__ h, long long n) {
    long long i = (long long)blockIdx.x * blockDim.x + threadIdx.x;
    if (i >= n) return;
    int col = (int)(i & (D - 1));
    float v = agg[i] * scale[col] + shift[col];
    h[i] = fmaxf(v, 0.0f) + h[i];
}

// ---------------------------------------------------------------------------
// global mean pool: one wave per node, float4 per lane
// ---------------------------------------------------------------------------
__global__ void gcn_pool_kernel(const float* __restrict__ h,
                                const int* __restrict__ batch,
                                float* __restrict__ psum,
                                float* __restrict__ pcnt, int N) {
    long long gw = ((long long)blockIdx.x * blockDim.x + threadIdx.x) >> 5;
    int lane = threadIdx.x & 31;
    if (gw >= N) return;
    int b = batch[gw];
    const float4 v = *(const float4*)(h + gw * D + lane * 4);
    float* p = psum + (long long)b * D + lane * 4;
    atomicAdd(p + 0, v.x);
    atomicAdd(p + 1, v.y);
    atomicAdd(p + 2, v.z);
    atomicAdd(p + 3, v.w);
    if (lane == 0) atomicAdd(&pcnt[b], 1.0f);
}

__global__ void gcn_pool_finalize_kernel(const float* __restrict__ psum,
                                         const float* __restrict__ pcnt,
                                         float* __restrict__ g, int n_graphs) {
    int i = blockIdx.x * blockDim.x + threadIdx.x;
    if (i >= n_graphs * D) return;
    int row = i / D;
    g[i] = psum[i] / fmaxf(pcnt[row], 1.0f);
}

// ---------------------------------------------------------------------------
// tiny MLP head layer: out[M x Nout] = act(in[M x K] @ W^T + b)
// ---------------------------------------------------------------------------
__global__ void gcn_mlp_kernel(const float* __restrict__ in,
                               const float* __restrict__ W,
                               const float* __restrict__ b,
                               float* __restrict__ out,
                               int M, int K, int Nout, int do_relu) {
    int i = blockIdx.x * blockDim.x + threadIdx.x;
    if (i >= M * Nout) return;
    int row = i / Nout, o = i % Nout;
    float acc = b[o];
    for (int k = 0; k < K; ++k) acc += in[row * K + k] * W[o * K + k];
    out[i] = do_relu ? fmaxf(acc, 0.0f) : acc;
}

// ---------------------------------------------------------------------------
// host orchestration
// ---------------------------------------------------------------------------
extern "C" void kernel_launch(void* const* d_in, const int* in_sizes, int n_in,
                              void* d_out, int out_size, void* d_ws, size_t ws_size,
                              hipStream_t stream) {
    const float* x     = (const float*)d_in[0];
    const int*   ei    = (const int*)  d_in[1];   // [2, E] flat: src then dst
    const int*   batch = (const int*)  d_in[2];
    const float* W_emb = (const float*)d_in[3];
    const float* b_emb = (const float*)d_in[4];
    const float* Wc    = (const float*)d_in[5];   // [4,128,128]
    // d_in[6] = bc: cancels exactly inside batch-norm (x - mean) -> unused
    const float* gamma = (const float*)d_in[7];   // [4,128]
    const float* beta  = (const float*)d_in[8];   // [4,128]
    const float* W1    = (const float*)d_in[9];
    const float* b1    = (const float*)d_in[10];
    const float* W2    = (const float*)d_in[11];
    const float* b2    = (const float*)d_in[12];
    const float* W3    = (const float*)d_in[13];
    const float* b3    = (const float*)d_in[14];

    const int N = in_sizes[0] / D;
    const int E = in_sizes[1] / 2;
    const int NG = 128;                      // N_GRAPHS
    const long long ND = (long long)N * D;

    const int* src = ei;
    const int* dst = ei + E;

    // workspace layout (floats)
    float* ws    = (float*)d_ws;
    float* h     = ws;                       // ND
    float* t     = h + ND;                   // ND
    float* agg   = t + ND;                   // ND
    float* deg   = agg + ND;                 // N
    float* dinv  = deg + N;                  // N
    float* stat  = dinv + N;                 // sums(128) sqs(128) scale(128) shift(128)
    float* sums  = stat;
    float* sqs   = stat + D;
    float* scale = stat + 2 * D;
    float* shift = stat + 3 * D;
    float* psum  = stat + 4 * D;             // NG*D
    float* pcnt  = psum + NG * D;            // NG
    float* g     = pcnt + NG;                // NG*D
    float* m1    = g + NG * D;               // NG*64
    float* m2    = m1 + NG * 64;             // NG*32

    const int ZB = 256;

    // ---- degree / normalization ----
    gcn_zero_kernel<<<1024, ZB, 0, stream>>>(deg, N);
    gcn_deg_kernel<<<(E + ZB - 1) / ZB, ZB, 0, stream>>>(dst, deg, E);
    gcn_dinv_kernel<<<(N + ZB - 1) / ZB, ZB, 0, stream>>>(deg, dinv, N);

    // ---- embedding GEMM: h = x @ W_emb^T + b_emb ----
    const int gemm_blocks = (N + 31) / 32;   // 32 rows per block
    gcn_gemm_wmma_kernel<<<gemm_blocks, 256, 0, stream>>>(x, W_emb, b_emb, h, N);

    // ---- 4 GCN layers ----
    const long long segs = (long long)E + N;               // edges + self-loops
    const int scat_blocks = (int)((segs * 32 + 255) / 256);
    for (int l = 0; l < 4; ++l) {
        gcn_gemm_wmma_kernel<<<gemm_blocks, 256, 0, stream>>>(h, Wc + (long long)l * D * D,
                                                              nullptr, t, N);
        gcn_zero_kernel<<<4096, ZB, 0, stream>>>(agg, ND);
        gcn_scatter_kernel<<<scat_blocks, 256, 0, stream>>>(t, src, dst, dinv, agg, E, N);

        gcn_zero_kernel<<<1, ZB, 0, stream>>>(sums, 2 * D);
        gcn_bn_stats_kernel<<<240, D, 0, stream>>>(agg, sums, sqs, N);
        gcn_bn_finalize_kernel<<<1, D, 0, stream>>>(sums, sqs, gamma + l * D, beta + l * D,
                                                    scale, shift, N);
        gcn_bn_apply_kernel<<<(int)((ND + 255) / 256), 256, 0, stream>>>(agg, scale, shift, h, ND);
    }

    // ---- mean pool over graphs ----
    gcn_zero_kernel<<<64, ZB, 0, stream>>>(psum, NG * D + NG);
    gcn_pool_kernel<<<(int)(((long long)N * 32 + 255) / 256), 256, 0, stream>>>(h, batch, psum, pcnt, N);
    gcn_pool_finalize_kernel<<<(NG * D + ZB - 1) / ZB, ZB, 0, stream>>>(psum, pcnt, g, NG);

    // ---- MLP head ----
    gcn_mlp_kernel<<<(NG * 64 + ZB - 1) / ZB, ZB, 0, stream>>>(g,  W1, b1, m1, NG, D,  64, 1);
    gcn_mlp_kernel<<<(NG * 32 + ZB - 1) / ZB, ZB, 0, stream>>>(m1, W2, b2, m2, NG, 64, 32, 1);
    gcn_mlp_kernel<<<(NG * 10 + ZB - 1) / ZB, ZB, 0, stream>>>(m2, W3, b3, (float*)d_out,
                                                               NG, 32, 10, 0);
}